// SkipGramModel_68917045232170
// MI455X (gfx1250) — compile-verified
//
#include <hip/hip_runtime.h>
#include <hip/hip_bf16.h>
#include <math.h>

#define VOCAB  100000
#define EMB    128
#define BPOS   16384
#define BNEG   81920
#define CTX    10
#define ROWS   (BPOS + BNEG)

#define WAVES  2            // waves per block
#define TPB    (WAVES * 32)
#define TILE   16           // rows per wave (one WMMA tile)
#define STRIDE 132          // padded LDS row stride in floats (128+4 -> conflict-free)

typedef float v2f __attribute__((ext_vector_type(2)));
typedef float v8f __attribute__((ext_vector_type(8)));

__device__ __forceinline__ float neg_log_sigmoid(float x) {
    // -log(sigmoid(x)) = log1p(exp(-|x|)) - min(x, 0)   (numerically stable)
    return log1pf(expf(-fabsf(x))) - fminf(x, 0.0f);
}

__global__ __launch_bounds__(TPB) void sg_partial_kernel(
    const int* __restrict__ pos_u, const int* __restrict__ pos_v,
    const int* __restrict__ neg_u, const int* __restrict__ neg_v,
    const float* __restrict__ w, float* __restrict__ partial)
{
    __shared__ __align__(16) float lA[WAVES][TILE][STRIDE]; // emb_u (context sums)
    __shared__ __align__(16) float lB[WAVES][TILE][STRIDE]; // emb_v (targets)
    __shared__ float wsum[WAVES];

    const int lane = threadIdx.x & 31;
    const int wid  = threadIdx.x >> 5;
    const int base = (blockIdx.x * WAVES + wid) * TILE;   // first row of this wave's tile

    // ---------- Phase 1: gather + context-sum into LDS ----------
    // One wave per row: lane l covers dims [4l, 4l+3] -> 512B row = 4 coalesced cachelines.
    const float4* w4 = reinterpret_cast<const float4*>(w);
    for (int r = 0; r < TILE; ++r) {
        const int row = base + r;
        const int* uidx;
        int vi;
        if (row < BPOS) { uidx = pos_u + row * CTX;            vi = pos_v[row]; }
        else            { int nr = row - BPOS;
                          uidx = neg_u + nr * CTX;             vi = neg_v[nr]; }
        float ax = 0.f, ay = 0.f, az = 0.f, aw = 0.f;
        #pragma unroll
        for (int c = 0; c < CTX; ++c) {
            const float4 t = w4[(size_t)uidx[c] * (EMB / 4) + lane];
            ax += t.x; ay += t.y; az += t.z; aw += t.w;
        }
        const float4 vv = w4[(size_t)vi * (EMB / 4) + lane];
        float* pa = &lA[wid][r][4 * lane];
        pa[0] = ax;   pa[1] = ay;   pa[2] = az;   pa[3] = aw;
        float* pb = &lB[wid][r][4 * lane];
        pb[0] = vv.x; pb[1] = vv.y; pb[2] = vv.z; pb[3] = vv.w;
    }
    __syncthreads();

    // ---------- Phase 2: 16 batched dot products = diag(A * B^T) via WMMA ----------
    // A 16x4 f32 layout: lane L holds A[L%16][kb + 2*(L/16) + j] in VGPR j.
    // B 4x16 layout with B[k][n] = embV[n][k] gives the identical address pattern.
    const int m  = lane & 15;
    const int hi = lane >> 4;
    const float* arow = &lA[wid][m][2 * hi];
    const float* brow = &lB[wid][m][2 * hi];
    v8f cacc = {0.f, 0.f, 0.f, 0.f, 0.f, 0.f, 0.f, 0.f};
    #pragma unroll
    for (int kb = 0; kb < EMB; kb += 4) {
        v2f a = *(const v2f*)(arow + kb);
        v2f b = *(const v2f*)(brow + kb);
        cacc = __builtin_amdgcn_wmma_f32_16x16x4_f32(
            /*neg_a=*/false, a, /*neg_b=*/false, b,
            /*c_mod=*/(short)0, cacc, /*reuse_a=*/false, /*reuse_b=*/false);
    }

    // C diag: (m,m) at lane m / VGPR m for m<8; lane m+16 / VGPR m-8 for m>=8.
    const int ci = (lane < 8) ? lane : (lane - 24);   // VGPR index (valid on active lanes)
    float e = cacc[0];
    #pragma unroll
    for (int j = 1; j < 8; ++j) e = (ci == j) ? cacc[j] : e;

    float contrib = 0.0f;
    if (lane < 8 || lane >= 24) {
        const int r   = (lane < 8) ? lane : (lane - 16);   // tile row owned by this lane
        const int row = base + r;
        const float x = (row < BPOS) ? e : -e;             // negatives: log_sigmoid(-score)
        contrib = neg_log_sigmoid(x);                      // already negated for final loss
    }

    // Deterministic wave tree-reduce.
    #pragma unroll
    for (int off = 16; off > 0; off >>= 1)
        contrib += __shfl_xor(contrib, off, 32);
    if (lane == 0) wsum[wid] = contrib;
    __syncthreads();
    if (threadIdx.x == 0) {
        float s = 0.f;
        #pragma unroll
        for (int i = 0; i < WAVES; ++i) s += wsum[i];
        partial[blockIdx.x] = s;
    }
}

__global__ __launch_bounds__(256) void sg_reduce_kernel(
    const float* __restrict__ partial, int n, float* __restrict__ out)
{
    __shared__ double s[256];
    double acc = 0.0;
    for (int i = threadIdx.x; i < n; i += 256) acc += (double)partial[i];
    s[threadIdx.x] = acc;
    __syncthreads();
    for (int stride = 128; stride > 0; stride >>= 1) {
        if ((int)threadIdx.x < stride) s[threadIdx.x] += s[threadIdx.x + stride];
        __syncthreads();
    }
    if (threadIdx.x == 0) out[0] = (float)s[0];
}

extern "C" void kernel_launch(void* const* d_in, const int* in_sizes, int n_in,
                              void* d_out, int out_size, void* d_ws, size_t ws_size,
                              hipStream_t stream) {
    const int*   pos_u = (const int*)d_in[0];
    const int*   pos_v = (const int*)d_in[1];
    const int*   neg_u = (const int*)d_in[2];
    const int*   neg_v = (const int*)d_in[3];
    const float* w     = (const float*)d_in[4];
    float* out     = (float*)d_out;
    float* partial = (float*)d_ws;              // 3072 floats = 12 KB of scratch

    const int nblocks = ROWS / (WAVES * TILE);  // 98304 / 32 = 3072 (exact)
    sg_partial_kernel<<<nblocks, TPB, 0, stream>>>(pos_u, pos_v, neg_u, neg_v, w, partial);
    sg_reduce_kernel<<<1, 256, 0, stream>>>(partial, nblocks, out);
}